// ContinuousMask_18056042512479
// MI455X (gfx1250) — compile-verified
//
#include <hip/hip_runtime.h>

// ContinuousMask: out[b,t] = 1.0f unless t is covered by any window [s, s+l)
// for s in starts[b, :], else 0.0f.
//
// Per-row difference array packed 2 positions/word (bias 0x2000 per halfword,
// no cross-halfword carry possible since per-position counts are bounded by
// nseg=8192 and a halfword is >=1 before any decrement), LDS ds_add atomics,
// conflict-free skewed block scan, float4 (b128) output stores.
//
// CDNA5 paths: the starts stream (64 MiB) is double-buffered through LDS with
// GLOBAL_LOAD_ASYNC_TO_LDS_B128 + s_wait_asynccnt.  Each lane consumes exactly
// the 16B its own async load wrote, so the pipeline is wave-private: no
// barriers, one tile always in flight (async loads complete in order).
//
// Roofline: ~64MB read + ~134MB write ~= 200MB -> ~8.6us floor at 23.3 TB/s.
// No matrix contraction anywhere => WMMA inapplicable.

#define T_DIM     16384
#define NTHREADS  256
#define CHUNK     (T_DIM / NTHREADS)      // 64 positions per thread
#define WPT       (CHUNK / 2)             // 32 packed words per thread
// skew(w) = w + (w>>5): thread t's word j lives at 33*t + j -> banks
// (33t + j) mod 64 distinct over 32 lanes => conflict-free scan passes.
// Max touched word index: skew(T/2) = 8448 (close at exactly T, never read).
#define DELTA_WORDS 8452
#define BIAS      8192
#define PACK_BIAS 0x20002000
#define TILE      (NTHREADS * 4)          // 1024 ints (4KB) per staging tile

#if defined(__has_builtin)
#if __has_builtin(__builtin_amdgcn_global_load_async_to_lds_b128)
#define USE_ASYNC_LDS 1
#endif
#endif

__device__ __forceinline__ int skw(int w) { return w + (w >> 5); }

#if defined(USE_ASYNC_LDS)
typedef int v4i_async __attribute__((vector_size(4 * sizeof(int))));
typedef __attribute__((address_space(1))) v4i_async* g_v4i_ptr;
typedef __attribute__((address_space(3))) v4i_async* l_v4i_ptr;

__device__ __forceinline__ void async_load_b128(const void* g, void* l) {
  __builtin_amdgcn_global_load_async_to_lds_b128(
      (g_v4i_ptr)g, (l_v4i_ptr)l, 0, 0);
}
#endif

__device__ __forceinline__ void scatter4(int* delta, int4 s, int l) {
  int e;
  atomicAdd(&delta[skw(s.x >> 1)], (s.x & 1) ? 65536 : 1);
  e = s.x + l;
  atomicAdd(&delta[skw(e  >> 1)], (e  & 1) ? -65536 : -1);
  atomicAdd(&delta[skw(s.y >> 1)], (s.y & 1) ? 65536 : 1);
  e = s.y + l;
  atomicAdd(&delta[skw(e  >> 1)], (e  & 1) ? -65536 : -1);
  atomicAdd(&delta[skw(s.z >> 1)], (s.z & 1) ? 65536 : 1);
  e = s.z + l;
  atomicAdd(&delta[skw(e  >> 1)], (e  & 1) ? -65536 : -1);
  atomicAdd(&delta[skw(s.w >> 1)], (s.w & 1) ? 65536 : 1);
  e = s.w + l;
  atomicAdd(&delta[skw(e  >> 1)], (e  & 1) ? -65536 : -1);
}

__global__ __launch_bounds__(NTHREADS)
void ContinuousMask_18056042512479_kernel(const int* __restrict__ starts,
                                          const int* __restrict__ lptr,
                                          float* __restrict__ out,
                                          int nseg) {
  __shared__ int delta[DELTA_WORDS];
  __shared__ int sums[NTHREADS];
#if defined(USE_ASYNC_LDS)
  __shared__ int stage[2 * TILE];
#endif

  const int row = blockIdx.x;
  const int tid = threadIdx.x;
  const int l   = lptr[0];

  // ---- init biased difference array ---------------------------------------
  #pragma unroll
  for (int i = tid; i < DELTA_WORDS; i += NTHREADS) delta[i] = PACK_BIAS;
  __syncthreads();

  // ---- scatter: +1 at s, -1 at s+l (packed halfword per position) ---------
  const int* rowp = starts + (size_t)row * nseg;
  int done = 0;

#if defined(USE_ASYNC_LDS)
  {
    const int ntiles = nseg / TILE;
    if (ntiles > 0) {
      const char* grow = (const char*)rowp;
      int* buf0 = stage + tid * 4;           // this lane's 16B slot, buffer 0
      int* buf1 = stage + TILE + tid * 4;    // this lane's 16B slot, buffer 1
      async_load_b128(grow + (size_t)tid * 16, buf0);
      for (int t = 0; t < ntiles; ++t) {
        int* cur = (t & 1) ? buf1 : buf0;
        int* nxt = (t & 1) ? buf0 : buf1;
        if (t + 1 < ntiles) {
          async_load_b128(grow + (size_t)(t + 1) * TILE * 4 + (size_t)tid * 16,
                          nxt);
          asm volatile("s_wait_asynccnt 0x1" ::: "memory");
        } else {
          asm volatile("s_wait_asynccnt 0x0" ::: "memory");
        }
        int4 s = *(const int4*)cur;          // wave-private: no barrier needed
        scatter4(delta, s, l);
      }
      done = ntiles * TILE;
    }
  }
#endif

  // remainder (and full path when async builtin unavailable)
  {
    const int4* rv = (const int4*)(rowp + done);
    const int   nv = (nseg - done) >> 2;
    for (int i = tid; i < nv; i += NTHREADS) {
      __builtin_prefetch((const void*)(rv + i + NTHREADS), 0, 1);
      scatter4(delta, rv[i], l);
    }
  }
  __syncthreads();

  // ---- pass 1: per-thread partial sum over its 64 positions (32 words) ----
  const int a = tid * (WPT + 1);             // skewed base: skw(tid*WPT)
  int local = 0;
  #pragma unroll
  for (int j = 0; j < WPT; ++j) {
    unsigned w = (unsigned)delta[a + j];
    local += (int)(w & 0xFFFFu) + (int)(w >> 16);
  }
  local -= CHUNK * BIAS;

  // ---- block-wide inclusive scan of 256 partials --------------------------
  sums[tid] = local;
  __syncthreads();
  #pragma unroll
  for (int off = 1; off < NTHREADS; off <<= 1) {
    int v = (tid >= off) ? sums[tid - off] : 0;
    __syncthreads();
    if (tid >= off) sums[tid] += v;
    __syncthreads();
  }
  int run = (tid > 0) ? sums[tid - 1] : 0;   // exclusive prefix for this chunk

  // ---- pass 2: running sum -> mask, float4 (b128) stores ------------------
  float4* ov = (float4*)(out + (size_t)row * T_DIM + tid * CHUNK);
  #pragma unroll
  for (int g = 0; g < CHUNK / 4; ++g) {
    unsigned w0 = (unsigned)delta[a + 2 * g];
    unsigned w1 = (unsigned)delta[a + 2 * g + 1];
    float4 f;
    run += (int)(w0 & 0xFFFFu) - BIAS;  f.x = (run > 0) ? 0.0f : 1.0f;
    run += (int)(w0 >> 16)     - BIAS;  f.y = (run > 0) ? 0.0f : 1.0f;
    run += (int)(w1 & 0xFFFFu) - BIAS;  f.z = (run > 0) ? 0.0f : 1.0f;
    run += (int)(w1 >> 16)     - BIAS;  f.w = (run > 0) ? 0.0f : 1.0f;
    ov[g] = f;
  }
}

extern "C" void kernel_launch(void* const* d_in, const int* in_sizes, int n_in,
                              void* d_out, int out_size, void* d_ws, size_t ws_size,
                              hipStream_t stream) {
  (void)d_ws; (void)ws_size; (void)n_in;
  const int* starts = (const int*)d_in[0];
  const int* lptr   = (const int*)d_in[2];   // inputs: starts, T, l
  float*     out    = (float*)d_out;

  const int B    = out_size / T_DIM;         // rows
  const int nseg = in_sizes[0] / B;          // windows per row

  if (B <= 0 || nseg <= 0) return;
  ContinuousMask_18056042512479_kernel<<<B, NTHREADS, 0, stream>>>(
      starts, lptr, out, nseg);
}